// GraphEmbedding_4320737100169
// MI455X (gfx1250) — compile-verified
//
#include <hip/hip_runtime.h>

#define DVEC 128
#define TILE_M 32
#define APAD 4            // LDS row pad: stride 132 words -> lane banks stride by 4, no conflicts

typedef __attribute__((ext_vector_type(2))) float v2f;
typedef __attribute__((ext_vector_type(8))) float v8f;

// ---------------- degree / norm setup ----------------
__global__ void k_init_deg(float* __restrict__ deg, int n) {
    int i = blockIdx.x * blockDim.x + threadIdx.x;
    if (i < n) deg[i] = 2.0f;   // improved=True self-loop weight 2
}

__global__ void k_count_deg(const int* __restrict__ dst, float* __restrict__ deg, int e) {
    int i = blockIdx.x * blockDim.x + threadIdx.x;
    if (i < e) atomicAdd(&deg[dst[i]], 1.0f);
}

__global__ void k_dinv(float* __restrict__ deg_dinv, float* __restrict__ self_norm, int n) {
    int i = blockIdx.x * blockDim.x + threadIdx.x;
    if (i < n) {
        float dv = rsqrtf(deg_dinv[i]);
        deg_dinv[i] = dv;                 // in-place: deg -> deg^-0.5
        self_norm[i] = 2.0f * dv * dv;
    }
}

__global__ void k_enorm(const int* __restrict__ src, const int* __restrict__ dst,
                        const float* __restrict__ dinv, float* __restrict__ enorm, int e) {
    int i = blockIdx.x * blockDim.x + threadIdx.x;
    if (i < e) enorm[i] = dinv[src[i]] * dinv[dst[i]];
}

// ---------------- expansion: h = log(x+1) @ We + be ----------------
__global__ void k_expand(const float* __restrict__ x, const float* __restrict__ We,
                         const float* __restrict__ be, float* __restrict__ h, int n) {
    __shared__ float lx[16];
    int node = blockIdx.x;
    int c = threadIdx.x;                  // 0..127
    if (c < 11) lx[c] = __logf(x[(size_t)node * 11 + c] + 1.0f);
    __syncthreads();
    float acc = be[c];
#pragma unroll
    for (int k = 0; k < 11; ++k) acc += lx[k] * We[k * DVEC + c];
    h[(size_t)node * DVEC + c] = acc;
}

// ---------------- GEMM t = h @ W, fused agg init (WMMA f32 16x16x4) ----------------
// block = 256 threads (8 waves). Block owns 32 rows staged to LDS via async-to-LDS.
// Wave w owns column tile w (16 cols) and computes TWO 16x16 M-subtiles, reusing
// each B fragment for two WMMAs (halves B vmem traffic per FLOP).
__global__ void k_gcn_gemm(const float* __restrict__ h, const float* __restrict__ W,
                           const float* __restrict__ bgl, const float* __restrict__ self_norm,
                           float* __restrict__ t, float* __restrict__ agg, int n) {
    __shared__ float As[TILE_M][DVEC + APAD];   // ~16.9 KB, padded rows
    int m0 = blockIdx.x * TILE_M;
    int tid = threadIdx.x;                      // 0..255

    // async stage of 32x128 A tile: 1024 b128 transfers, 4 per thread, ASYNCcnt-tracked
    unsigned lds_base = (unsigned)(size_t)&As[0][0];
    for (int i = tid; i < TILE_M * (DVEC / 4); i += 256) {
        int r  = i >> 5;                        // 32 float4 per row
        int c4 = i & 31;
        int row = m0 + r;
        if (row >= n) row = n - 1;              // clamp (stores are guarded later)
        const float* gp = h + (size_t)row * DVEC + (c4 << 2);
        unsigned ldst = lds_base + (unsigned)(r * (DVEC + APAD) + (c4 << 2)) * 4u;
        asm volatile("global_load_async_to_lds_b128 %0, %1, off"
                     :: "v"(ldst), "v"(gp) : "memory");
    }
    asm volatile("s_wait_asynccnt 0" ::: "memory");
    __syncthreads();

    int wave = tid >> 5;                        // 0..7 -> column tile
    int lane = tid & 31;
    int hl    = lane & 15;                      // M (A) / N (B,C,D) within half-wave
    int kh    = (lane >> 4) << 1;               // A/B K base: 0 or 2
    int rbase = (lane >> 4) << 3;               // C/D row base: 0 or 8
    int col   = (wave << 4) + hl;

    v8f acc0 = {};
    v8f acc1 = {};
    for (int k = 0; k < DVEC; k += 4) {
        v2f b, a0, a1;
        b.x  = W[(size_t)(k + kh) * DVEC + col];
        b.y  = W[(size_t)(k + kh + 1) * DVEC + col];
        a0.x = As[hl][k + kh];
        a0.y = As[hl][k + kh + 1];
        a1.x = As[16 + hl][k + kh];
        a1.y = As[16 + hl][k + kh + 1];
        acc0 = __builtin_amdgcn_wmma_f32_16x16x4_f32(
            false, a0, false, b, (short)0, acc0, false, false);
        acc1 = __builtin_amdgcn_wmma_f32_16x16x4_f32(
            false, a1, false, b, (short)0, acc1, false, false);
    }

    float bgv = bgl[col];
#pragma unroll
    for (int j = 0; j < 8; ++j) {
        int row0 = m0 + rbase + j;
        if (row0 < n) {
            float v = acc0[j];
            size_t off = (size_t)row0 * DVEC + col;
            t[off]   = v;
            agg[off] = self_norm[row0] * v + bgv;    // self-loop + bias, fused
        }
        int row1 = m0 + 16 + rbase + j;
        if (row1 < n) {
            float v = acc1[j];
            size_t off = (size_t)row1 * DVEC + col;
            t[off]   = v;
            agg[off] = self_norm[row1] * v + bgv;
        }
    }
}

// ---------------- edge scatter: agg[dst] += enorm * t[src] ----------------
// one wave32 per edge, one float4 per lane (L2-resident traffic + f32 atomics)
__global__ void k_edge_scatter(const int* __restrict__ src, const int* __restrict__ dst,
                               const float* __restrict__ enorm, const float* __restrict__ t,
                               float* __restrict__ agg, int e) {
    int gid  = blockIdx.x * blockDim.x + threadIdx.x;
    int edge = gid >> 5;
    int lane = gid & 31;
    if (edge >= e) return;
    int s = src[edge];
    int d = dst[edge];
    float w = enorm[edge];
    float4 v = ((const float4*)(t + (size_t)s * DVEC))[lane];
    float* ap = agg + (size_t)d * DVEC + lane * 4;
    atomicAdd(ap + 0, v.x * w);
    atomicAdd(ap + 1, v.y * w);
    atomicAdd(ap + 2, v.z * w);
    atomicAdd(ap + 3, v.w * w);
}

// ---------------- residual + relu ----------------
__global__ void k_relu_residual(float* __restrict__ h, const float* __restrict__ agg,
                                unsigned int total) {
    unsigned int i = blockIdx.x * blockDim.x + threadIdx.x;
    if (i < total) {
        float a = agg[i];
        h[i] += a > 0.0f ? a : 0.0f;
    }
}

// ---------------- global add pool ----------------
__global__ void k_zero(float* __restrict__ p, unsigned int total) {
    unsigned int i = blockIdx.x * blockDim.x + threadIdx.x;
    if (i < total) p[i] = 0.0f;
}

__global__ void k_pool(const float* __restrict__ h, const int* __restrict__ batch,
                       float* __restrict__ out, int n) {
    int gid  = blockIdx.x * blockDim.x + threadIdx.x;
    int node = gid >> 5;
    int lane = gid & 31;
    if (node >= n) return;
    int g = batch[node];
    float4 v = ((const float4*)(h + (size_t)node * DVEC))[lane];
    float* op = out + (size_t)g * DVEC + lane * 4;
    atomicAdd(op + 0, v.x);
    atomicAdd(op + 1, v.y);
    atomicAdd(op + 2, v.z);
    atomicAdd(op + 3, v.w);
}

extern "C" void kernel_launch(void* const* d_in, const int* in_sizes, int n_in,
                              void* d_out, int out_size, void* d_ws, size_t ws_size,
                              hipStream_t stream) {
    const float* x     = (const float*)d_in[0];
    const int*   edge  = (const int*)d_in[1];
    const int*   batch = (const int*)d_in[2];
    // d_in[3] = num_graphs (device scalar) — recovered from out_size instead
    const float* We    = (const float*)d_in[4];
    const float* be    = (const float*)d_in[5];
    const float* Wg    = (const float*)d_in[6];
    const float* bg    = (const float*)d_in[7];
    float* out = (float*)d_out;

    int n = in_sizes[0] / 11;                     // 100000
    int e = in_sizes[1] / 2;                      // 400000
    int n_layers = in_sizes[6] / (DVEC * DVEC);   // 5
    const int* srcp = edge;
    const int* dstp = edge + e;

    // workspace carve (fits 192MB L2: h/t/agg = 3*51.2MB)
    char* ws = (char*)d_ws;
    float* h    = (float*)ws; ws += (size_t)n * DVEC * sizeof(float);
    float* t    = (float*)ws; ws += (size_t)n * DVEC * sizeof(float);
    float* agg  = (float*)ws; ws += (size_t)n * DVEC * sizeof(float);
    float* dinv = (float*)ws; ws += (size_t)n * sizeof(float);
    float* sn   = (float*)ws; ws += (size_t)n * sizeof(float);
    float* en   = (float*)ws; ws += (size_t)e * sizeof(float);

    k_init_deg<<<(n + 255) / 256, 256, 0, stream>>>(dinv, n);
    k_count_deg<<<(e + 255) / 256, 256, 0, stream>>>(dstp, dinv, e);
    k_dinv<<<(n + 255) / 256, 256, 0, stream>>>(dinv, sn, n);
    k_enorm<<<(e + 255) / 256, 256, 0, stream>>>(srcp, dstp, dinv, en, e);
    k_expand<<<n, DVEC, 0, stream>>>(x, We, be, h, n);

    int gemm_blocks = (n + TILE_M - 1) / TILE_M;
    unsigned int total = (unsigned int)n * DVEC;
    unsigned int escatter_blocks = (unsigned int)(((long long)e * 32 + 255) / 256);
    for (int l = 0; l < n_layers; ++l) {
        k_gcn_gemm<<<gemm_blocks, 256, 0, stream>>>(
            h, Wg + (size_t)l * DVEC * DVEC, bg + (size_t)l * DVEC, sn, t, agg, n);
        k_edge_scatter<<<escatter_blocks, 256, 0, stream>>>(srcp, dstp, en, t, agg, e);
        k_relu_residual<<<(total + 255) / 256, 256, 0, stream>>>(h, agg, total);
    }

    unsigned int out_total = (unsigned int)out_size;
    k_zero<<<(out_total + 255) / 256, 256, 0, stream>>>(out, out_total);
    unsigned int pool_blocks = (unsigned int)(((long long)n * 32 + 255) / 256);
    k_pool<<<pool_blocks, 256, 0, stream>>>(h, batch, out, n);
}